// Prediction_24094766531304
// MI455X (gfx1250) — compile-verified
//
#include <hip/hip_runtime.h>
#include <math.h>

#define H   512
#define S   256
#define BB  32
#define R   16
#define INW 4
#define NNW 16
#define NW  20          // IN + NN
#define OPW 6
#define NEGV (-1000000000000.0f)

typedef float v2f __attribute__((ext_vector_type(2)));
typedef float v8f __attribute__((ext_vector_type(8)));

#if __has_builtin(__builtin_amdgcn_tanhf)
__device__ __forceinline__ float fast_tanh(float x) { return __builtin_amdgcn_tanhf(x); }
#elif __has_builtin(__builtin_amdgcn_tanh_f32)
__device__ __forceinline__ float fast_tanh(float x) { return __builtin_amdgcn_tanh_f32(x); }
#else
__device__ __forceinline__ float fast_tanh(float x) { return tanhf(x); }
#endif

__device__ __forceinline__ float waveReduceSum(float v) {
#pragma unroll
    for (int off = 16; off > 0; off >>= 1) v += __shfl_xor(v, off, 32);
    return v;
}

__device__ __forceinline__ float sigmoidf(float x) { return 1.f / (1.f + expf(-x)); }

// ---------------------------------------------------------------------------
// f32 WMMA fragment layouts (V_WMMA_F32_16X16X4_F32):
//   A 16x4 : lanes 0-15 -> M=lane, K={0,1}; lanes 16-31 -> M=lane-16, K={2,3}
//   B 4x16 : lanes 0-15 -> N=lane, K={0,1}; lanes 16-31 -> N=lane-16, K={2,3}
//   C/D    : vgpr v, lanes 0-15 -> (m0+v, n0+lane); lanes 16-31 -> (m0+v+8, ...)
// ---------------------------------------------------------------------------

// Big-tile GEMM: one wave computes a 32x64 tile (2 M-subtiles x 4 N-subtiles).
// grid = (N/64, M/32).  ~10.7 FLOP/B vs 4 for the 16x16 version.
__global__ void gemm_wmma_f32_big(const float* __restrict__ A, int lda,
                                  const float* __restrict__ Bm, int ldb,
                                  float* __restrict__ C, int ldc,
                                  int K, const float* __restrict__ bias, int act)
{
    const int lane = threadIdx.x;
    const int m0   = blockIdx.y << 5;
    const int n0   = blockIdx.x << 6;
    const int half = lane >> 4;
    const int l15  = lane & 15;

    const float* aRow0 = A + (long long)(m0 + l15) * lda + (half << 1);
    const float* aRow1 = aRow0 + (long long)16 * lda;
    const float* bBase = Bm + (long long)(half << 1) * ldb + (n0 + l15);

    v8f acc[2][4];
#pragma unroll
    for (int i = 0; i < 2; ++i)
#pragma unroll
        for (int j = 0; j < 4; ++j)
            acc[i][j] = (v8f){0.f, 0.f, 0.f, 0.f, 0.f, 0.f, 0.f, 0.f};

    for (int k = 0; k < K; k += 4) {
        v2f a0, a1, b[4];
        a0.x = aRow0[k]; a0.y = aRow0[k + 1];
        a1.x = aRow1[k]; a1.y = aRow1[k + 1];
        const float* bp = bBase + (long long)k * ldb;
#pragma unroll
        for (int j = 0; j < 4; ++j) {
            b[j].x = bp[j * 16];
            b[j].y = bp[j * 16 + ldb];
        }
#pragma unroll
        for (int j = 0; j < 4; ++j) {
            acc[0][j] = __builtin_amdgcn_wmma_f32_16x16x4_f32(false, a0, false, b[j],
                                                              (short)0, acc[0][j], false, false);
            acc[1][j] = __builtin_amdgcn_wmma_f32_16x16x4_f32(false, a1, false, b[j],
                                                              (short)0, acc[1][j], false, false);
        }
    }

    const int drBase = m0 + (half << 3);
#pragma unroll
    for (int i = 0; i < 2; ++i) {
        const int dr = drBase + i * 16;
#pragma unroll
        for (int j = 0; j < 4; ++j) {
            const int dc = n0 + j * 16 + l15;
            const float bv = bias ? bias[dc] : 0.f;
#pragma unroll
            for (int v = 0; v < 8; ++v) {
                float val = acc[i][j][v] + bv;
                if (act) val = fast_tanh(val);
                C[(long long)(dr + v) * ldc + dc] = val;
            }
        }
    }
}

// Small/batched GEMM: one wave per 16x16 tile; grid = (N/16, M/16, batch)
__global__ void gemm_wmma_f32(const float* __restrict__ A, int lda, long long sA,
                              const float* __restrict__ Bm, int ldb, long long sB,
                              float* __restrict__ C, int ldc, long long sC,
                              int K, const float* __restrict__ bias, int act)
{
    const int z = blockIdx.z;
    A  += (long long)z * sA;
    Bm += (long long)z * sB;
    C  += (long long)z * sC;

    const int lane = threadIdx.x;
    const int m0   = blockIdx.y << 4;
    const int n0   = blockIdx.x << 4;
    const int half = lane >> 4;
    const int l15  = lane & 15;

    const float* aRow = A  + (long long)(m0 + l15) * lda + (half << 1);
    const float* bCol = Bm + (long long)(half << 1) * ldb + (n0 + l15);

    v8f acc = {0.f, 0.f, 0.f, 0.f, 0.f, 0.f, 0.f, 0.f};
    for (int k = 0; k < K; k += 4) {
        v2f a, b;
        a.x = aRow[k];
        a.y = aRow[k + 1];
        b.x = bCol[(long long)k * ldb];
        b.y = bCol[(long long)(k + 1) * ldb];
        acc = __builtin_amdgcn_wmma_f32_16x16x4_f32(false, a, false, b,
                                                    (short)0, acc, false, false);
    }

    const int dr = m0 + (half << 3);
    const int dc = n0 + l15;
    const float bv = bias ? bias[dc] : 0.f;
#pragma unroll
    for (int v = 0; v < 8; ++v) {
        float val = acc[v] + bv;
        if (act) val = fast_tanh(val);
        C[(long long)(dr + v) * ldc + dc] = val;
    }
}

// ---------------------------------------------------------------------------
// Gated node update: current_node = has_left ? out_r : out_l
// ---------------------------------------------------------------------------
__global__ void k_gates(const float* __restrict__ node_emb, const float* __restrict__ left,
                        const int* __restrict__ has_left,
                        const float* __restrict__ Wcl,  const float* __restrict__ bcl,
                        const float* __restrict__ Wclg, const float* __restrict__ bclg,
                        const float* __restrict__ Wcr,  const float* __restrict__ bcr,
                        const float* __restrict__ Wcrg, const float* __restrict__ bcrg,
                        float* __restrict__ out_cn)
{
    const int b = blockIdx.y;
    const int h = blockIdx.x * 64 + threadIdx.x;
    const float* c = node_emb + b * H;
    const float* l = left + b * H;

    float s1 = bcl[h], s2 = bclg[h];
    for (int k = 0; k < H; ++k) {
        float cv = c[k];
        s1 = fmaf(cv, Wcl[k * H + h], s1);
        s2 = fmaf(cv, Wclg[k * H + h], s2);
    }
    float out_l = fast_tanh(s1) * sigmoidf(s2);

    float s3 = bcr[h], s4 = bcrg[h];
    for (int k = 0; k < 2 * H; ++k) {
        float lv = (k < H) ? l[k] : c[k - H];
        s3 = fmaf(lv, Wcr[k * H + h], s3);
        s4 = fmaf(lv, Wcrg[k * H + h], s4);
    }
    float out_r = fast_tanh(s3) * sigmoidf(s4);

    out_cn[b * H + h] = has_left[b] ? out_r : out_l;
}

// en[b,s] = sum_h tanh(E[(s,b),h] + cnWh[b,h] + battn[h]) * wattn_s[h] + battn_s
__global__ void k_en(const float* __restrict__ E, const float* __restrict__ cnWh,
                     const float* __restrict__ battn, const float* __restrict__ wattn_s,
                     const float* __restrict__ battn_s, float* __restrict__ en)
{
    int gid  = blockIdx.x * (blockDim.x >> 5) + (threadIdx.x >> 5);
    int lane = threadIdx.x & 31;
    int s = gid & (S - 1);
    int b = gid >> 8;
    const float4* e  = (const float4*)(E + (long long)(s * BB + b) * H);
    const float4* cw = (const float4*)(cnWh + b * H);
    const float4* ba = (const float4*)battn;
    const float4* wa = (const float4*)wattn_s;
    float acc = 0.f;
#pragma unroll
    for (int i = lane; i < H / 4; i += 32) {
        float4 ev = e[i], cv = cw[i], bv = ba[i], wv = wa[i];
        acc += fast_tanh(ev.x + cv.x + bv.x) * wv.x
             + fast_tanh(ev.y + cv.y + bv.y) * wv.y
             + fast_tanh(ev.z + cv.z + bv.z) * wv.z
             + fast_tanh(ev.w + cv.w + bv.w) * wv.w;
    }
    acc = waveReduceSum(acc);
    if (lane == 0) en[gid] = acc + battn_s[0];
}

// masked softmax over S -> current_attn
__global__ void k_softmax_attn(const float* __restrict__ en, const int* __restrict__ seq_mask,
                               float* __restrict__ attn)
{
    __shared__ float red[S];
    int b = blockIdx.x, s = threadIdx.x;
    float v = seq_mask[b * S + s] ? NEGV : en[b * S + s];
    red[s] = v; __syncthreads();
    for (int off = S / 2; off > 0; off >>= 1) {
        if (s < off) red[s] = fmaxf(red[s], red[s + off]);
        __syncthreads();
    }
    float m = red[0]; __syncthreads();
    float e = expf(v - m);
    red[s] = e; __syncthreads();
    for (int off = S / 2; off > 0; off >>= 1) {
        if (s < off) red[s] += red[s + off];
        __syncthreads();
    }
    attn[b * S + s] = e / red[0];
}

// current_context[b,h] = sum_s attn[b,s] * eo[(s,b),h]
__global__ void k_ctx(const float* __restrict__ attn, const float* __restrict__ eo,
                      float* __restrict__ cc)
{
    __shared__ float a[S];
    int b = blockIdx.y;
    int h = blockIdx.x * 256 + threadIdx.x;
    a[threadIdx.x] = attn[b * S + threadIdx.x];
    __syncthreads();
    float acc = 0.f;
    for (int s = 0; s < S; ++s)
        acc = fmaf(a[s], eo[(long long)(s * BB + b) * H + h], acc);
    cc[b * H + h] = acc;
}

// fen[(b*R+r)*S+s] = sum_h tanh(E2[(s,b),h] + F[r,h] + bfattn[h]) * wfattn_s[h] + bfattn_s
__global__ void k_fen(const float* __restrict__ E2, const float* __restrict__ F,
                      const float* __restrict__ bfattn, const float* __restrict__ wfattn_s,
                      const float* __restrict__ bfattn_s, float* __restrict__ fen)
{
    int gid  = blockIdx.x * (blockDim.x >> 5) + (threadIdx.x >> 5);
    int lane = threadIdx.x & 31;
    int s = gid & (S - 1);
    int r = (gid >> 8) & (R - 1);
    int b = gid >> 12;
    const float4* e  = (const float4*)(E2 + (long long)(s * BB + b) * H);
    const float4* f  = (const float4*)(F + r * H);
    const float4* ba = (const float4*)bfattn;
    const float4* wa = (const float4*)wfattn_s;
    float acc = 0.f;
#pragma unroll
    for (int i = lane; i < H / 4; i += 32) {
        float4 ev = e[i], fv = f[i], bv = ba[i], wv = wa[i];
        acc += fast_tanh(ev.x + fv.x + bv.x) * wv.x
             + fast_tanh(ev.y + fv.y + bv.y) * wv.y
             + fast_tanh(ev.z + fv.z + bv.z) * wv.z
             + fast_tanh(ev.w + fv.w + bv.w) * wv.w;
    }
    acc = waveReduceSum(acc);
    if (lane == 0) fen[gid] = acc + bfattn_s[0];
}

// comb = softmax_S(masked fen) + 0.2*current_attn
__global__ void k_softmax_comb(const float* __restrict__ fen, const int* __restrict__ seq_mask,
                               const float* __restrict__ attn, float* __restrict__ comb)
{
    __shared__ float red[S];
    int br = blockIdx.x;          // b*R + r
    int b  = br >> 4;
    int s  = threadIdx.x;
    float v = seq_mask[b * S + s] ? NEGV : fen[br * S + s];
    red[s] = v; __syncthreads();
    for (int off = S / 2; off > 0; off >>= 1) {
        if (s < off) red[s] = fmaxf(red[s], red[s + off]);
        __syncthreads();
    }
    float m = red[0]; __syncthreads();
    float e = expf(v - m);
    red[s] = e; __syncthreads();
    for (int off = S / 2; off > 0; off >>= 1) {
        if (s < off) red[s] += red[s + off];
        __syncthreads();
    }
    comb[br * S + s] = e / red[0] + 0.2f * attn[b * S + s];
}

// formula_prob: logits[b,r] = ff[(b,r),:]·Wfp2 + bfp2 ; softmax over R
__global__ void k_fprob(const float* __restrict__ ff, const float* __restrict__ Wfp2,
                        const float* __restrict__ bfp2, float* __restrict__ fp)
{
    __shared__ float lg[R];
    int b    = blockIdx.x;
    int r    = threadIdx.x >> 5;
    int lane = threadIdx.x & 31;
    const float4* row = (const float4*)(ff + (long long)(b * R + r) * H);
    const float4* w4  = (const float4*)Wfp2;
    float acc = 0.f;
#pragma unroll
    for (int i = lane; i < H / 4; i += 32) {
        float4 rv = row[i], wv = w4[i];
        acc += rv.x * wv.x + rv.y * wv.y + rv.z * wv.z + rv.w * wv.w;
    }
    acc = waveReduceSum(acc);
    if (lane == 0) lg[r] = acc + bfp2[0];
    __syncthreads();
    if (threadIdx.x < R) {
        float m = -INFINITY;
        for (int i = 0; i < R; ++i) m = fmaxf(m, lg[i]);
        float sum = 0.f;
        for (int i = 0; i < R; ++i) sum += expf(lg[i] - m);
        fp[b * R + threadIdx.x] = expf(lg[threadIdx.x] - m) / sum;
    }
}

// assemble emb_w output and ein = [cn | cc | emb_w]
__global__ void k_ein(const float* __restrict__ cn, const float* __restrict__ cc,
                      const float* __restrict__ embw, const float* __restrict__ pades,
                      float* __restrict__ out_emb, float* __restrict__ ein)
{
    int row = blockIdx.x;         // b*NW + n
    int b = row / NW, n = row % NW;
    for (int j = threadIdx.x; j < 3 * H; j += blockDim.x) {
        float v;
        if (j < H)          v = cn[b * H + j];
        else if (j < 2 * H) v = cc[b * H + j - H];
        else {
            int h = j - 2 * H;
            v = (n < INW) ? embw[n * H + h]
                          : pades[(long long)(b * NNW + (n - INW)) * H + h];
            out_emb[(long long)row * H + h] = v;
        }
        ein[(long long)row * 3 * H + j] = v;
    }
}

// num_score[b,n] = mask ? NEG : nst[(b,n),:]·wscore_s
__global__ void k_nscore(const float* __restrict__ nst, const float* __restrict__ wscore_s,
                         const int* __restrict__ mask_nums, float* __restrict__ ns)
{
    int gid  = blockIdx.x * (blockDim.x >> 5) + (threadIdx.x >> 5);   // 640 rows
    int lane = threadIdx.x & 31;
    const float4* row = (const float4*)(nst + (long long)gid * H);
    const float4* w4  = (const float4*)wscore_s;
    float acc = 0.f;
#pragma unroll
    for (int i = lane; i < H / 4; i += 32) {
        float4 rv = row[i], wv = w4[i];
        acc += rv.x * wv.x + rv.y * wv.y + rv.z * wv.z + rv.w * wv.w;
    }
    acc = waveReduceSum(acc);
    if (lane == 0) ns[gid] = mask_nums[gid] ? NEGV : acc;
}

// op[b,o] = leaf[b,:]·Wops[:,o] + bops[o]
__global__ void k_op(const float* __restrict__ cn, const float* __restrict__ cc,
                     const float* __restrict__ Wops, const float* __restrict__ bops,
                     float* __restrict__ op)
{
    int b    = blockIdx.x;
    int o    = threadIdx.x >> 5;   // 6 waves
    int lane = threadIdx.x & 31;
    float acc = 0.f;
    for (int k = lane; k < 2 * H; k += 32) {
        float lv = (k < H) ? cn[b * H + k] : cc[b * H + k - H];
        acc = fmaf(lv, Wops[k * OPW + o], acc);
    }
    acc = waveReduceSum(acc);
    if (lane == 0) op[b * OPW + o] = acc + bops[o];
}

// ---------------------------------------------------------------------------
extern "C" void kernel_launch(void* const* d_in, const int* in_sizes, int n_in,
                              void* d_out, int out_size, void* d_ws, size_t ws_size,
                              hipStream_t stream)
{
    (void)in_sizes; (void)n_in; (void)out_size; (void)ws_size;

    const float* node_emb = (const float*)d_in[0];
    const float* left     = (const float*)d_in[1];
    const float* eo       = (const float*)d_in[2];   // (S,B,H)
    const float* pades    = (const float*)d_in[3];
    const float* fscores  = (const float*)d_in[4];
    const int*   has_left = (const int*)  d_in[5];
    const int*   seq_mask = (const int*)  d_in[6];
    const int*   mask_nums= (const int*)  d_in[7];
    const float* embw     = (const float*)d_in[8];
    const float* Wcl      = (const float*)d_in[9];
    const float* bcl      = (const float*)d_in[10];
    const float* Wclg     = (const float*)d_in[11];
    const float* bclg     = (const float*)d_in[12];
    const float* Wcr      = (const float*)d_in[13];
    const float* bcr      = (const float*)d_in[14];
    const float* Wcrg     = (const float*)d_in[15];
    const float* bcrg     = (const float*)d_in[16];
    const float* Wops     = (const float*)d_in[17];
    const float* bops     = (const float*)d_in[18];
    const float* Wattn    = (const float*)d_in[19];
    const float* battn    = (const float*)d_in[20];
    const float* wattn_s  = (const float*)d_in[21];
    const float* battn_s  = (const float*)d_in[22];
    const float* Wfattn   = (const float*)d_in[23];
    const float* bfattn   = (const float*)d_in[24];
    const float* wfattn_s = (const float*)d_in[25];
    const float* bfattn_s = (const float*)d_in[26];
    const float* Wscore   = (const float*)d_in[27];
    const float* bscore   = (const float*)d_in[28];
    const float* wscore_s = (const float*)d_in[29];
    const float* Wfp1     = (const float*)d_in[30];
    const float* bfp1     = (const float*)d_in[31];
    const float* Wfp2     = (const float*)d_in[32];
    const float* bfp2     = (const float*)d_in[33];

    float* out = (float*)d_out;
    float* out_ns  = out;                 // (B,NW)      640
    float* out_op  = out + 640;           // (B,OPW)     192
    float* out_cn  = out + 832;           // (B,1,H)   16384
    float* out_cc  = out + 17216;         // (B,1,H)   16384
    float* out_emb = out + 33600;         // (B,NW,H) 327680
    float* out_fp  = out + 361280;        // (B,R)       512

    float* ws = (float*)d_ws;
    float* ws_E    = ws;                  // 8192*512 (reused for E1 then E2)
    float* ws_cnWh = ws + 4194304;        // 32*512
    float* ws_F    = ws + 4210688;        // 16*512
    float* ws_en   = ws + 4218880;        // 32*256
    float* ws_attn = ws + 4227072;        // 32*256
    float* ws_fen  = ws + 4235264;        // 32*16*256
    float* ws_comb = ws + 4366336;        // 32*16*256
    float* ws_fc   = ws + 4497408;        // 32*16*512
    float* ws_ff   = ws + 4759552;        // 32*16*512
    float* ws_ein  = ws + 5022208;        // 640*1536
    float* ws_nst  = ws + 6005248;        // 640*512

    const int SB = S * BB;   // 8192

    auto gemmBig = [&](const float* A, int lda, const float* Bm, int ldb,
                       float* C, int ldc, int M, int N, int K,
                       const float* bias, int act) {
        dim3 g(N / 64, M / 32, 1);
        gemm_wmma_f32_big<<<g, 32, 0, stream>>>(A, lda, Bm, ldb, C, ldc, K, bias, act);
    };
    auto gemmSm = [&](const float* A, int lda, long long sA,
                      const float* Bm, int ldb, long long sB,
                      float* C, int ldc, long long sC,
                      int M, int N, int K, int batch,
                      const float* bias, int act) {
        dim3 g(N / 16, M / 16, batch);
        gemm_wmma_f32<<<g, 32, 0, stream>>>(A, lda, sA, Bm, ldb, sB, C, ldc, sC, K, bias, act);
    };

    // 1) gated node update -> current_node
    k_gates<<<dim3(H / 64, BB), 64, 0, stream>>>(node_emb, left, has_left,
                                                 Wcl, bcl, Wclg, bclg,
                                                 Wcr, bcr, Wcrg, bcrg, out_cn);

    // 2) cnWh = current_node @ Wh  (Wh = Wattn rows 0..H-1)
    gemmBig(out_cn, H, Wattn, H, ws_cnWh, H, BB, H, H, nullptr, 0);

    // 3) E1 = encoder_outputs @ We  (We = Wattn rows H..2H-1)
    gemmBig(eo, H, Wattn + H * H, H, ws_E, H, SB, H, H, nullptr, 0);

    // 4) en reduction + 5) masked softmax + 6) current_context
    k_en<<<SB / 8, 256, 0, stream>>>(ws_E, ws_cnWh, battn, wattn_s, battn_s, ws_en);
    k_softmax_attn<<<BB, S, 0, stream>>>(ws_en, seq_mask, ws_attn);
    k_ctx<<<dim3(H / 256, BB), 256, 0, stream>>>(ws_attn, eo, out_cc);

    // 7) E2 = encoder_outputs @ Wfe (reuse ws_E)  8) F = formula_scores @ Wff
    gemmBig(eo, H, Wfattn, H, ws_E, H, SB, H, H, nullptr, 0);
    gemmSm(fscores, H, 0, Wfattn + H * H, H, 0, ws_F, H, 0, R, H, H, 1, nullptr, 0);

    // 9) fen reduction + 10) softmax + 0.2*current_attn
    k_fen<<<(BB * R * S) / 8, 256, 0, stream>>>(ws_E, ws_F, bfattn, wfattn_s, bfattn_s, ws_fen);
    k_softmax_comb<<<BB * R, S, 0, stream>>>(ws_fen, seq_mask, ws_attn, ws_comb);

    // 11) formulas_context: batched (16x256)@(256x512) over b, B-matrix is eo strided
    gemmSm(ws_comb, S, (long long)R * S,
           eo, BB * H, (long long)H,
           ws_fc, H, (long long)R * H,
           R, H, S, BB, nullptr, 0);

    // 12) formula_feat = tanh(fc @ Wfp1 + bfp1)   13) formula_prob
    gemmBig(ws_fc, H, Wfp1, H, ws_ff, H, BB * R, H, H, bfp1, 1);
    k_fprob<<<BB, 512, 0, stream>>>(ws_ff, Wfp2, bfp2, out_fp);

    // 14) emb_w output + ein assembly
    k_ein<<<BB * NW, 256, 0, stream>>>(out_cn, out_cc, embw, pades, out_emb, ws_ein);

    // 15) nst = tanh(ein @ Wscore + bscore)   16) num_score
    gemmBig(ws_ein, 3 * H, Wscore, H, ws_nst, H, BB * NW, H, 3 * H, bscore, 1);
    k_nscore<<<(BB * NW) / 8, 256, 0, stream>>>(ws_nst, wscore_s, mask_nums, out_ns);

    // 17) op = leaf_input @ Wops + bops
    k_op<<<BB, OPW * 32, 0, stream>>>(out_cn, out_cc, Wops, bops, out_op);
}